// PoseEstimator_44143673869124
// MI455X (gfx1250) — compile-verified
//
#include <hip/hip_runtime.h>
#include <math.h>

#define NPARTS 2
#define NSRC   24576
#define STEP   3
#define M      8192            // sampled points per part (24576/3)
#define TILE   16
#define ROWS_PER_BLOCK 128     // 8 waves x 16 rows
#define COLS_PER_BLOCK 2048    // 4 column strips over M
#define THREADS 256
#define UNROLL 4

typedef float v2f __attribute__((ext_vector_type(2)));
typedef float v8f __attribute__((ext_vector_type(8)));

__device__ __forceinline__ void make_transform(const float* q4, const float* t3, float T[4][4]) {
    float a = q4[0], b = q4[1], c = q4[2], d = q4[3];
    float n = sqrtf(a*a + b*b + c*c + d*d);
    a /= n; b /= n; c /= n; d /= n;
    T[0][0] = 1.f - 2.f*c*c - 2.f*d*d; T[0][1] = 2.f*b*c - 2.f*a*d;       T[0][2] = 2.f*a*c + 2.f*b*d;       T[0][3] = t3[0];
    T[1][0] = 2.f*b*c + 2.f*a*d;       T[1][1] = 1.f - 2.f*b*b - 2.f*d*d; T[1][2] = 2.f*c*d - 2.f*a*b;       T[1][3] = t3[1];
    T[2][0] = 2.f*b*d - 2.f*a*c;       T[2][1] = 2.f*a*b + 2.f*c*d;       T[2][2] = 1.f - 2.f*b*b - 2.f*c*c; T[2][3] = t3[2];
    T[3][0] = 0.f; T[3][1] = 0.f; T[3][2] = 0.f; T[3][3] = 1.f;
}

// Kernel 1: build transforms; pack A' = (-2x,-2y,-2z, |a|^2) so the K=4 WMMA
// directly yields (|a|^2 - 2 a.b); B' = (u,v,w,1) with cn = |b|^2 separate.
// Init min arrays to +inf; write transforms to out[2..33].
__global__ void prep_kernel(const float* __restrict__ cam, const float* __restrict__ cad,
                            const float* __restrict__ quat, const float* __restrict__ tra,
                            float* __restrict__ A, float* __restrict__ B,
                            float* __restrict__ cn,
                            unsigned* __restrict__ rowMin, unsigned* __restrict__ colMin,
                            float* __restrict__ out) {
    int gid = blockIdx.x * blockDim.x + threadIdx.x;
    if (gid >= NPARTS * M) return;
    int p = gid / M;
    int i = gid - p * M;

    float T[4][4];
    make_transform(quat + p * 4, tra + p * 3, T);

    size_t src = ((size_t)p * NSRC + (size_t)i * STEP) * 3;
    float x = cad[src + 0], y = cad[src + 1], z = cad[src + 2];
    float tx = T[0][0]*x + T[0][1]*y + T[0][2]*z + T[0][3];
    float ty = T[1][0]*x + T[1][1]*y + T[1][2]*z + T[1][3];
    float tz = T[2][0]*x + T[2][1]*y + T[2][2]*z + T[2][3];
    A[gid*4+0] = -2.0f*tx; A[gid*4+1] = -2.0f*ty; A[gid*4+2] = -2.0f*tz;
    A[gid*4+3] = tx*tx + ty*ty + tz*tz;          // row norm rides in the K=4 slot

    float cx = cam[src + 0], cy = cam[src + 1], cz = cam[src + 2];
    B[gid*4+0] = cx; B[gid*4+1] = cy; B[gid*4+2] = cz; B[gid*4+3] = 1.0f;
    cn[gid] = cx*cx + cy*cy + cz*cz;

    rowMin[gid] = 0x7F800000u;   // +inf bits
    colMin[gid] = 0x7F800000u;

    if (gid < NPARTS * 16) {     // block 0 only: emit transforms (2 x 4 x 4)
        int pp = gid / 16, idx = gid & 15, r = idx >> 2, cc = idx & 3;
        float Tt[4][4];
        make_transform(quat + pp * 4, tra + pp * 3, Tt);
        out[2 + pp * 16 + idx] = Tt[r][cc];
    }
}

// Kernel 2: fused chamfer.  Grid = (rowStrips=64, colStrips=4, parts=2) -> 512
// blocks / 4096 waves.  The block's 2048-column B/cn strip (40 KB) is staged
// once into LDS.  Phase-split body: 4 tiles of DS loads (lowered to
// ds_load_2addr_*), 4 back-to-back v_wmma_f32_16x16x4_f32, then epilogues.
// Epilogue adds are expressed as within-tile v2f pairs + broadcast |b|^2 so
// they lower to v_pk_add_f32 on the contiguous WMMA D registers (no moves).
//   d2 = max(wmma(A',B') + |b|^2, 0)
// Col mins accumulate in LDS via ds_min_u32, row mins in registers; both merge
// to global with uint atomicMin (valid ordering for non-negative floats).
__global__ void __launch_bounds__(THREADS)
chamfer_kernel(const float* __restrict__ A, const float* __restrict__ B,
               const float* __restrict__ cn,
               unsigned* __restrict__ rowMin, unsigned* __restrict__ colMin) {
    __shared__ unsigned lcol[COLS_PER_BLOCK];   // 8 KB
    __shared__ float4   Bs[COLS_PER_BLOCK];     // 32 KB: (u,v,w,1) per column
    __shared__ float    cns[COLS_PER_BLOCK];    // 8 KB: |b|^2 per column

    const int tid  = threadIdx.x;
    const int part = blockIdx.z;
    const int colStripBase = blockIdx.y * COLS_PER_BLOCK;

    const float4* __restrict__ Bv  = (const float4*)(B + (size_t)part * M * 4);
    const float*  __restrict__ cnp = cn + (size_t)part * M;

    for (int k = tid; k < COLS_PER_BLOCK; k += THREADS) {
        Bs[k]   = Bv[colStripBase + k];
        cns[k]  = cnp[colStripBase + k];
        lcol[k] = 0x7F800000u;
    }
    __syncthreads();

    const int wave = tid >> 5;
    const int lane = tid & 31;
    const int half = lane >> 4;     // K-half for A/B operands
    const int m16  = lane & 15;     // row (A) / col (B,C) within tile
    const int rowBase = blockIdx.x * ROWS_PER_BLOCK + wave * TILE;

    const float* Ap = A + (size_t)part * M * 4;

    // A operand: lane L holds row (L%16), K = {2*(L/16), 2*(L/16)+1}
    v2f a = *(const v2f*)(Ap + (size_t)(rowBase + m16) * 4 + 2 * half);

    float rowRun[8];
#pragma unroll
    for (int v = 0; v < 8; ++v) rowRun[v] = __builtin_inff();

    for (int ct = 0; ct < COLS_PER_BLOCK / TILE; ct += UNROLL) {
        // Phase 1: issue all LDS loads (distinct destination registers).
        v2f   b[UNROLL];
        float cnv[UNROLL];
#pragma unroll
        for (int u = 0; u < UNROLL; ++u) {
            const int col = (ct + u) * TILE + m16;
            b[u]   = *(const v2f*)((const float*)&Bs[col] + 2 * half);
            cnv[u] = cns[col];
        }
        // Phase 2: 4 independent WMMAs back to back.
        v8f c[UNROLL];
#pragma unroll
        for (int u = 0; u < UNROLL; ++u) {
            v8f z = {};
            c[u] = __builtin_amdgcn_wmma_f32_16x16x4_f32(
                       false, a, false, b[u], (short)0, z, false, false);
        }
        // Phase 3: epilogues + LDS col-min atomics.  Adds are within-tile v2f
        // pairs (contiguous D VGPRs) + broadcast cn -> v_pk_add_f32, no moves.
#pragma unroll
        for (int u = 0; u < UNROLL; ++u) {
            v2f cc;
            cc[0] = cnv[u]; cc[1] = cnv[u];
            float d2s[8];
#define CHAMFER_PAIR(VP)                                                        \
            {                                                                   \
                v2f pair = __builtin_shufflevector(c[u], c[u],                  \
                                                   2*(VP), 2*(VP) + 1);         \
                v2f s = pair + cc;                                              \
                d2s[2*(VP)]     = fmaxf(s[0], 0.0f);                            \
                d2s[2*(VP) + 1] = fmaxf(s[1], 0.0f);                            \
            }
            CHAMFER_PAIR(0)
            CHAMFER_PAIR(1)
            CHAMFER_PAIR(2)
            CHAMFER_PAIR(3)
#undef CHAMFER_PAIR
            float mv = __builtin_inff();
#pragma unroll
            for (int v = 0; v < 8; ++v) {
                rowRun[v] = fminf(rowRun[v], d2s[v]);
                mv = fminf(mv, d2s[v]);
            }
            atomicMin(&lcol[(ct + u) * TILE + m16], __float_as_uint(mv));
        }
    }

    // row mins: reduce across the 16 lanes of each half, then merge the 4
    // column-strip blocks via global atomicMin.
#pragma unroll
    for (int v = 0; v < 8; ++v) {
        float r = rowRun[v];
        r = fminf(r, __shfl_xor(r, 8, 32));
        r = fminf(r, __shfl_xor(r, 4, 32));
        r = fminf(r, __shfl_xor(r, 2, 32));
        r = fminf(r, __shfl_xor(r, 1, 32));
        if (m16 == 0)
            atomicMin(&rowMin[(size_t)part * M + rowBase + v + 8 * half],
                      __float_as_uint(r));
    }

    __syncthreads();
    for (int k = tid; k < COLS_PER_BLOCK; k += THREADS)
        atomicMin(&colMin[(size_t)part * M + colStripBase + k], lcol[k]);
}

// Kernel 3: mean(sqrt(min d2)) both directions, weight, kinematic term, objective.
__global__ void finalize_kernel(const unsigned* __restrict__ rowMin,
                                const unsigned* __restrict__ colMin,
                                const float* __restrict__ pw,
                                const float* __restrict__ ja,
                                float* __restrict__ out) {
    __shared__ float red[THREADS];
    __shared__ float tot[4];
    int tid = threadIdx.x;
    float acc[4] = {0.f, 0.f, 0.f, 0.f};
    for (int k = tid; k < M; k += THREADS) {
        for (int p = 0; p < NPARTS; ++p) {
            acc[2*p+0] += sqrtf(__uint_as_float(rowMin[(size_t)p * M + k]));
            acc[2*p+1] += sqrtf(__uint_as_float(colMin[(size_t)p * M + k]));
        }
    }
    for (int q = 0; q < 4; ++q) {
        red[tid] = acc[q];
        __syncthreads();
        for (int s = THREADS / 2; s > 0; s >>= 1) {
            if (tid < s) red[tid] += red[tid + s];
            __syncthreads();
        }
        if (tid == 0) tot[q] = red[0];
        __syncthreads();
    }
    if (tid == 0) {
        float e0 = (tot[0] + tot[1]) / (float)M;
        float e1 = (tot[2] + tot[3]) / (float)M;
        float e_geo = pw[0] * e0 + pw[1] * e1;
        // diff = T0 @ J0^T - T1 @ J1^T  (4x2), Frobenius norm
        float ss = 0.f;
        for (int r = 0; r < 4; ++r)
            for (int c = 0; c < 2; ++c) {
                float s0 = 0.f, s1 = 0.f;
                for (int k = 0; k < 4; ++k) {
                    s0 += out[2 + 0*16 + r*4 + k] * ja[0*8 + c*4 + k];
                    s1 += out[2 + 1*16 + r*4 + k] * ja[1*8 + c*4 + k];
                }
                float d = s0 - s1;
                ss += d * d;
            }
        float fro  = sqrtf(ss);
        float ekin = 1.f / (1.f + expf(5.f * fro));   // sigmoid(-5*fro)
        out[1] = ekin;
        out[0] = e_geo + pw[NPARTS] * ekin;
    }
}

extern "C" void kernel_launch(void* const* d_in, const int* in_sizes, int n_in,
                              void* d_out, int out_size, void* d_ws, size_t ws_size,
                              hipStream_t stream) {
    const float* cam  = (const float*)d_in[0];   // camera_pts (2,24576,3)
    const float* cad  = (const float*)d_in[1];   // cad_pts    (2,24576,3)
    const float* pw   = (const float*)d_in[2];   // part_weight (3,)
    const float* quat = (const float*)d_in[3];   // rot_quat   (2,4)
    const float* tra  = (const float*)d_in[4];   // tra        (2,3,1)
    const float* ja   = (const float*)d_in[5];   // joint_axes (2,2,4)
    float* out = (float*)d_out;                  // [obj, e_kin, transforms(32)]

    float* ws = (float*)d_ws;                    // ~704 KB used
    float* A  = ws;                              // 2*8192*4
    float* B  = A  + (size_t)NPARTS * M * 4;     // 2*8192*4
    float* cn = B  + (size_t)NPARTS * M * 4;     // 2*8192
    unsigned* rowMin = (unsigned*)(cn + (size_t)NPARTS * M);  // 2*8192
    unsigned* colMin = rowMin + (size_t)NPARTS * M;           // 2*8192

    prep_kernel<<<dim3((NPARTS * M + THREADS - 1) / THREADS), THREADS, 0, stream>>>(
        cam, cad, quat, tra, A, B, cn, rowMin, colMin, out);
    chamfer_kernel<<<dim3(M / ROWS_PER_BLOCK, M / COLS_PER_BLOCK, NPARTS), THREADS, 0, stream>>>(
        A, B, cn, rowMin, colMin);
    finalize_kernel<<<1, THREADS, 0, stream>>>(rowMin, colMin, pw, ja, out);
}